// DynamicLIF_73538430042798
// MI455X (gfx1250) — compile-verified
//
#include <hip/hip_runtime.h>

// DynamicLIF on MI455X (gfx1250, wave32).
// 2 workgroups per batch element (64 blocks x 1024 threads): membrane state is
// 64 floats/thread -> fits the ~128 VGPR/wave budget of a 32-wave workgroup
// (round 1 spilled at 128 floats/thread). x streamed in / spikes streamed out
// once with non-temporal hints (256 MB @ 23.3 TB/s ~= 11 us roofline); 64 WGPs
// are needed to actually reach that bandwidth. Per-step cross-block exchange of
// channel sums goes through d_ws with release/acquire flags (parity
// double-buffered). The 32x128 matvec runs on V_WMMA_F32_16X16X4_F32.

typedef float v2f __attribute__((ext_vector_type(2)));
typedef float f4  __attribute__((ext_vector_type(4)));
typedef float v8f __attribute__((ext_vector_type(8)));

#define NB    32          // batch
#define NTS   8           // timesteps
#define NC    128         // channels
#define NHW   1024        // H*W
#define NHWH  512         // H*W per half-block
#define NCR   32          // C/4 (reduced dim)

// d_ws layout:
//   [0 .. 64KB)   float xsum[NB][2(half)][2(parity)][NC]
//   [64KB .. )    int   flags[NB*2]
#define WS_XSUM_FLOATS (NB * 2 * 2 * NC)
#define WS_FLAG_OFF    (WS_XSUM_FLOATS * sizeof(float))

__global__ void lif_init_flags(int* flags) {
    if (threadIdx.x < NB * 2) flags[threadIdx.x] = 0;
}

__global__ __launch_bounds__(1024)
void dynamic_lif_kernel(const float* __restrict__ x,
                        const float* __restrict__ w1,
                        const float* __restrict__ b1,
                        const float* __restrict__ w2,
                        const float* __restrict__ b2,
                        float* __restrict__ out,
                        float* __restrict__ xsum,
                        int*   __restrict__ flags)
{
    __shared__ float s_w1[NCR * NC];   // 32x128 = 16 KB
    __shared__ float s_w2[NCR];
    __shared__ float s_b1[NCR];
    __shared__ float s_xmean[NC];
    __shared__ float s_b2;

    const int tid  = threadIdx.x;
    const int lane = tid & 31;
    const int wv   = tid >> 5;             // wave id 0..31
    const int b    = blockIdx.x >> 1;      // batch id
    const int hb   = blockIdx.x & 1;       // which hw-half this block owns

    // ---- stage tiny weights into LDS once ----
    for (int i = tid; i < NCR * NC; i += 1024) s_w1[i] = w1[i];
    if (tid < NCR) { s_w2[tid] = w2[tid]; s_b1[tid] = b1[tid]; }
    if (tid == 0)  s_b2 = b2[0];
    __syncthreads();

    const int c0   = wv * 4;               // this wave's 4 channels
    const int half = lane >> 4;            // WMMA lane-half
    const int lsub = lane & 15;

    // membrane state: 4 channels x 4 float4 chunks = 64 floats in VGPRs
    f4 mem[4][4];
    #pragma unroll
    for (int i = 0; i < 4; ++i)
        #pragma unroll
        for (int j = 0; j < 4; ++j)
            mem[i][j] = (f4)0.0f;

    float tau = 0.5f;                      // TAU0

    for (int t = 0; t < NTS; ++t) {
        const size_t base = (((size_t)b * NTS + t) * NC) * (size_t)NHW
                          + (size_t)hb * NHWH;
        const size_t pf   = (t + 1 < NTS) ? (size_t)NC * NHW : 0;  // clamp: no peel
        float psum[4] = {0.f, 0.f, 0.f, 0.f};

        // ---- streaming pass: mem update, channel-sum, spike out, reset ----
        #pragma unroll
        for (int i = 0; i < 4; ++i) {
            const size_t cb = base + (size_t)(c0 + i) * NHW;
            #pragma unroll
            for (int j = 0; j < 4; ++j) {
                const size_t off = cb + (size_t)j * 128 + (size_t)lane * 4;
                __builtin_prefetch(x + off + pf, 0, 0);   // warm next timestep
                f4 xv = __builtin_nontemporal_load((const f4*)(x + off));
                f4 m  = mem[i][j];
                f4 sp;
                #pragma unroll
                for (int e = 0; e < 4; ++e) {
                    float mv = fmaf(m[e], tau, xv[e]);
                    psum[i] += mv;                           // pre-reset mean
                    float s  = (mv > 1.0f) ? 1.0f : 0.0f;    // zif(mem-THRESH)
                    sp[e]    = s;
                    m[e]     = (mv > 1.0f) ? 0.0f : mv;      // (1-spike)*mem
                }
                __builtin_nontemporal_store(sp, (f4*)(out + off));
                mem[i][j] = m;
            }
        }

        // ---- per-channel partial sums: wave32 xor-reduce, publish to ws ----
        const int par = t & 1;             // parity double-buffer
        #pragma unroll
        for (int i = 0; i < 4; ++i) {
            float v = psum[i];
            #pragma unroll
            for (int msk = 16; msk >= 1; msk >>= 1)
                v += __shfl_xor(v, msk, 32);
            if (lane == 0)
                xsum[(((b * 2 + hb) * 2 + par) * NC) + c0 + i] = v;
        }
        __syncthreads();                   // all partial sums issued

        // ---- release our flag, acquire partner's ----
        if (tid == 0) {
            __threadfence();               // push sums to device scope
            __hip_atomic_store(&flags[b * 2 + hb], t + 1,
                               __ATOMIC_RELEASE, __HIP_MEMORY_SCOPE_AGENT);
            while (__hip_atomic_load(&flags[b * 2 + (1 - hb)],
                                     __ATOMIC_ACQUIRE, __HIP_MEMORY_SCOPE_AGENT)
                   < t + 1)
                __builtin_amdgcn_s_sleep(1);
        }
        __syncthreads();                   // acquire visible block-wide

        // ---- combine both halves into the per-batch mean ----
        if (tid < NC) {
            float a = xsum[(((b * 2 + 0) * 2 + par) * NC) + tid];
            float c = xsum[(((b * 2 + 1) * 2 + par) * NC) + tid];
            s_xmean[tid] = (a + c) * (1.0f / (float)NHW);
        }
        __syncthreads();

        // ---- MLP via V_WMMA_F32_16X16X4_F32 (every wave, redundant) ----
        // A = w1 tile (16x4 f32): lanes0-15 hold K=0,1; lanes16-31 K=2,3.
        // B = x_mean replicated into all 16 columns -> every D column = matvec.
        v8f acc0 = (v8f)0.0f;   // rows 0..15 of w1
        v8f acc1 = (v8f)0.0f;   // rows 16..31 of w1
        #pragma unroll 4
        for (int k = 0; k < NC / 4; ++k) {
            const int k0 = k * 4 + 2 * half;
            v2f bv; bv.x = s_xmean[k0];               bv.y = s_xmean[k0 + 1];
            v2f a0; a0.x = s_w1[lsub * NC + k0];      a0.y = s_w1[lsub * NC + k0 + 1];
            v2f a1; a1.x = s_w1[(16 + lsub) * NC + k0];
                    a1.y = s_w1[(16 + lsub) * NC + k0 + 1];
            acc0 = __builtin_amdgcn_wmma_f32_16x16x4_f32(
                       false, a0, false, bv, (short)0, acc0, false, false);
            acc1 = __builtin_amdgcn_wmma_f32_16x16x4_f32(
                       false, a1, false, bv, (short)0, acc1, false, false);
        }

        // D layout: vgpr i, lanes0-15 -> M=i; lanes16-31 -> M=8+i.
        // This lane holds rows {half*8+i} (acc0) and {16+half*8+i} (acc1);
        // the two lane-halves cover complementary row sets -> one xor-add.
        float dot = 0.0f;
        #pragma unroll
        for (int i = 0; i < 8; ++i) {
            const int m0 = half * 8 + i;
            float e0 = acc0[i] + s_b1[m0];
            float e1 = acc1[i] + s_b1[m0 + 16];
            e0 = e0 > 0.0f ? e0 : 0.0f;                 // relu
            e1 = e1 > 0.0f ? e1 : 0.0f;
            dot = fmaf(s_w2[m0],      e0, dot);
            dot = fmaf(s_w2[m0 + 16], e1, dot);
        }
        dot += __shfl_xor(dot, 16, 32);                 // combine lane halves
        tau  = 1.0f / (1.0f + __expf(-(dot + s_b2)));   // sigmoid -> next tau
        // s_xmean is rewritten only after the NEXT step's post-spin barrier,
        // so no extra barrier is needed here.
    }
}

extern "C" void kernel_launch(void* const* d_in, const int* in_sizes, int n_in,
                              void* d_out, int out_size, void* d_ws, size_t ws_size,
                              hipStream_t stream) {
    const float* x  = (const float*)d_in[0];
    const float* w1 = (const float*)d_in[1];
    const float* b1 = (const float*)d_in[2];
    const float* w2 = (const float*)d_in[3];
    const float* b2 = (const float*)d_in[4];
    float* out = (float*)d_out;
    float* xsum  = (float*)d_ws;
    int*   flags = (int*)((char*)d_ws + WS_FLAG_OFF);
    (void)in_sizes; (void)n_in; (void)out_size; (void)ws_size;

    lif_init_flags<<<dim3(1), dim3(64), 0, stream>>>(flags);
    dynamic_lif_kernel<<<dim3(NB * 2), dim3(1024), 0, stream>>>(
        x, w1, b1, w2, b2, out, xsum, flags);
}